// GATReceptionPredictor_41558103556530
// MI455X (gfx1250) — compile-verified
//
#include <hip/hip_runtime.h>
#include <hip/hip_bf16.h>

typedef __attribute__((ext_vector_type(2))) float v2f;
typedef __attribute__((ext_vector_type(8))) float v8f;

#define HD 128
#define NHEAD 4
#define NEG_SLOPE 0.2f

// ---------- helpers ----------
__device__ __forceinline__ float lrelu(float v) { return v > 0.f ? v : NEG_SLOPE * v; }

// order-preserving float->uint key so GLOBAL_ATOMIC_MAX_U32 implements float max
__device__ __forceinline__ unsigned f2ord(float f) {
    unsigned u = __float_as_uint(f);
    return (u & 0x80000000u) ? ~u : (u | 0x80000000u);
}
__device__ __forceinline__ float ord2f(unsigned u) {
    return (u & 0x80000000u) ? __uint_as_float(u & 0x7fffffffu) : __uint_as_float(~u);
}

// ---------- zero fill ----------
__global__ void zero_f32(float* __restrict__ p, long long n) {
    long long i = (long long)blockIdx.x * blockDim.x + threadIdx.x;
    if (i < n) p[i] = 0.f;
}

// ---------- fp32 WMMA GEMM:  Y[N,128] = X[N,K] @ W[K,128] + bias, optional relu ----------
// One block = 16 rows. 8 waves, wave w owns output columns [16w,16w+16).
// A frag (16x4 f32): lanes 0-15 rows M=0..15 K={k,k+1}; lanes 16-31 same rows K={k+2,k+3}.
// B frag (4x16 f32): lanes 0-15 cols N=0..15 K={k,k+1}; lanes 16-31 same cols K={k+2,k+3}.
// C/D (16x16 f32): VGPR j -> row j (lanes 0-15) / row 8+j (lanes 16-31), col = lane&15.
template <int K>
__global__ void gemm128_wmma(const float* __restrict__ X, const float* __restrict__ W,
                             const float* __restrict__ bias, float* __restrict__ Y,
                             int Nrows, int act) {
    __shared__ float Xs[16 * K];
    const int base = blockIdx.x * 16;
    const int tid  = threadIdx.x;

    for (int i = tid; i < 16 * K; i += 256) {
        int r = i / K, c = i - r * K;
        int gr = base + r;
        Xs[i] = (gr < Nrows) ? X[(long long)gr * K + c] : 0.f;
    }
    __syncthreads();

    const int w     = tid >> 5;          // column tile 0..7
    const int l     = tid & 31;
    const int ln    = l & 15;
    const int hi    = l >> 4;            // 0 or 1
    const int khalf = hi << 1;           // 0 or 2
    const int ncol  = w * 16 + ln;

    v8f acc = {};
#pragma unroll
    for (int kk = 0; kk < K; kk += 4) {
        v2f a, b;
        a.x = Xs[ln * K + kk + khalf];
        a.y = Xs[ln * K + kk + khalf + 1];
        b.x = W[(kk + khalf) * HD + ncol];
        b.y = W[(kk + khalf + 1) * HD + ncol];
        acc = __builtin_amdgcn_wmma_f32_16x16x4_f32(false, a, false, b,
                                                    (short)0, acc, false, false);
    }

    const float bv    = bias[ncol];
    const int   rbase = base + (hi << 3);
#pragma unroll
    for (int j = 0; j < 8; ++j) {
        int row = rbase + j;
        if (row < Nrows) {
            float v = acc[j] + bv;
            if (act) v = fmaxf(v, 0.f);
            Y[(long long)row * HD + ncol] = v;
        }
    }
}

// ---------- edge pass 1: score[e,h] = sum_c lrelu(xl[s]+xr[d])*att ; segment max ----------
__global__ void edge_scores(const float* __restrict__ xl, const float* __restrict__ xr,
                            const int* __restrict__ ei, int E, int Etot,
                            const float* __restrict__ att,
                            float* __restrict__ sbuf, unsigned* __restrict__ smax) {
    int e = blockIdx.x * 8 + (threadIdx.x >> 5);
    if (e >= Etot) return;
    int l = threadIdx.x & 31;
    int s, d;
    if (e < E) { s = ei[e]; d = ei[E + e]; } else { s = e - E; d = s; }

    const float4 a = *(const float4*)(xl + (long long)s * HD + l * 4);
    const float4 b = *(const float4*)(xr + (long long)d * HD + l * 4);
    const float4 t = *(const float4*)(att + l * 4);

    float p = lrelu(a.x + b.x) * t.x + lrelu(a.y + b.y) * t.y +
              lrelu(a.z + b.z) * t.z + lrelu(a.w + b.w) * t.w;
    // reduce across the 8 lanes of one head
    p += __shfl_xor(p, 1, 32);
    p += __shfl_xor(p, 2, 32);
    p += __shfl_xor(p, 4, 32);
    if ((l & 7) == 0) {
        int h = l >> 3;
        sbuf[(long long)e * NHEAD + h] = p;
        atomicMax(&smax[d * NHEAD + h], f2ord(p));
    }
}

// ---------- edge pass 2: ex = exp(score - smax[d]); denom[d] += ex ----------
__global__ void edge_exp(const int* __restrict__ ei, int E, int Etot,
                         const unsigned* __restrict__ smax,
                         float* __restrict__ sbuf, float* __restrict__ denom) {
    long long i = (long long)blockIdx.x * 256 + threadIdx.x;
    if (i >= (long long)Etot * NHEAD) return;
    int e = (int)(i >> 2), h = (int)(i & 3);
    int d = (e < E) ? ei[E + e] : e - E;
    float m  = ord2f(smax[d * NHEAD + h]);
    float ex = expf(sbuf[i] - m);
    sbuf[i] = ex;
    atomicAdd(&denom[d * NHEAD + h], ex);
}

// ---------- edge pass 3: alpha = ex/denom ; accum[d] += xl[s] * alpha ----------
__global__ void edge_aggregate(const float* __restrict__ xl, const int* __restrict__ ei,
                               int E, int Etot,
                               const float* __restrict__ sbuf, const float* __restrict__ denom,
                               float* __restrict__ alpha_out, float* __restrict__ accum) {
    int e = blockIdx.x * 8 + (threadIdx.x >> 5);
    if (e >= Etot) return;
    int l = threadIdx.x & 31;
    int s, d;
    if (e < E) { s = ei[e]; d = ei[E + e]; } else { s = e - E; d = s; }

    int h = l >> 3;
    float al = sbuf[(long long)e * NHEAD + h] / denom[d * NHEAD + h];
    if ((l & 7) == 0) alpha_out[(long long)e * NHEAD + h] = al;

    const float4 a = *(const float4*)(xl + (long long)s * HD + l * 4);
    float* o = accum + (long long)d * HD + l * 4;
    atomicAdd(o + 0, a.x * al);
    atomicAdd(o + 1, a.y * al);
    atomicAdd(o + 2, a.z * al);
    atomicAdd(o + 3, a.w * al);
}

// ---------- finalize: accum += bias, optional relu ----------
__global__ void finalize_nodes(float* __restrict__ accum, const float* __restrict__ bo,
                               long long total, int act) {
    long long i = (long long)blockIdx.x * 256 + threadIdx.x;
    if (i >= total) return;
    float v = accum[i] + bo[i & (HD - 1)];
    if (act) v = fmaxf(v, 0.f);
    accum[i] = v;
}

// ---------- MLP head: probs[n] = sigmoid(Z[n,:] . W2 + b2) ----------
__global__ void mlp_out(const float* __restrict__ Z, const float* __restrict__ W2,
                        const float* __restrict__ b2, float* __restrict__ probs, int N) {
    int n = blockIdx.x * 8 + (threadIdx.x >> 5);
    if (n >= N) return;
    int l = threadIdx.x & 31;
    const float4 z = *(const float4*)(Z + (long long)n * HD + l * 4);
    const float4 w = *(const float4*)(W2 + l * 4);
    float p = z.x * w.x + z.y * w.y + z.z * w.z + z.w * w.w;
    p += __shfl_xor(p, 16, 32);
    p += __shfl_xor(p, 8, 32);
    p += __shfl_xor(p, 4, 32);
    p += __shfl_xor(p, 2, 32);
    p += __shfl_xor(p, 1, 32);
    if (l == 0) probs[n] = 1.f / (1.f + expf(-(p + b2[0])));
}

// ---------- orchestration ----------
extern "C" void kernel_launch(void* const* d_in, const int* in_sizes, int n_in,
                              void* d_out, int out_size, void* d_ws, size_t ws_size,
                              hipStream_t stream) {
    const float* x     = (const float*)d_in[0];
    const int*   ei    = (const int*)d_in[1];
    const float* enc_W = (const float*)d_in[4];
    const float* enc_b = (const float*)d_in[5];
    const float* Wl[2]  = {(const float*)d_in[6],  (const float*)d_in[12]};
    const float* bl[2]  = {(const float*)d_in[7],  (const float*)d_in[13]};
    const float* Wr[2]  = {(const float*)d_in[8],  (const float*)d_in[14]};
    const float* br[2]  = {(const float*)d_in[9],  (const float*)d_in[15]};
    const float* att[2] = {(const float*)d_in[10], (const float*)d_in[16]};
    const float* bo[2]  = {(const float*)d_in[11], (const float*)d_in[17]};
    const float* mW1 = (const float*)d_in[18];
    const float* mb1 = (const float*)d_in[19];
    const float* mW2 = (const float*)d_in[20];
    const float* mb2 = (const float*)d_in[21];

    const int N    = in_sizes[0] / 16;
    const int E    = in_sizes[1] / 2;
    const int Etot = E + N;

    // workspace layout
    float*    A     = (float*)d_ws;                         // N*128
    float*    B     = A + (size_t)N * HD;                   // N*128
    float*    C     = B + (size_t)N * HD;                   // N*128
    float*    sbuf  = C + (size_t)N * HD;                   // Etot*4
    unsigned* smax  = (unsigned*)(sbuf + (size_t)Etot * NHEAD); // N*4
    float*    denom = (float*)(smax + (size_t)N * NHEAD);   // N*4

    float* out      = (float*)d_out;
    float* alpha[2] = {out + N, out + N + (size_t)Etot * NHEAD};

    const int gemmGrid  = (N + 15) / 16;
    const int edgeGrid  = (Etot + 7) / 8;
    const long long nhd = (long long)N * HD;

    // encoder: A = x @ enc_W + enc_b
    gemm128_wmma<16><<<gemmGrid, 256, 0, stream>>>(x, enc_W, enc_b, A, N, 0);

    for (int L = 0; L < 2; ++L) {
        // B = xl, C = xr  (from A)
        gemm128_wmma<128><<<gemmGrid, 256, 0, stream>>>(A, Wl[L], bl[L], B, N, 0);
        gemm128_wmma<128><<<gemmGrid, 256, 0, stream>>>(A, Wr[L], br[L], C, N, 0);
        // A becomes the aggregation target
        zero_f32<<<(unsigned)((nhd + 255) / 256), 256, 0, stream>>>(A, nhd);
        zero_f32<<<(unsigned)((2LL * N * NHEAD + 255) / 256), 256, 0, stream>>>(
            (float*)smax, 2LL * N * NHEAD);  // zeros smax(u32 keys) + denom contiguously
        edge_scores<<<edgeGrid, 256, 0, stream>>>(B, C, ei, E, Etot, att[L], sbuf, smax);
        edge_exp<<<(unsigned)(((long long)Etot * NHEAD + 255) / 256), 256, 0, stream>>>(
            ei, E, Etot, smax, sbuf, denom);
        edge_aggregate<<<edgeGrid, 256, 0, stream>>>(B, ei, E, Etot, sbuf, denom,
                                                     alpha[L], A);
        finalize_nodes<<<(unsigned)((nhd + 255) / 256), 256, 0, stream>>>(
            A, bo[L], nhd, L == 0 ? 1 : 0);
    }

    // MLP: B = relu(A @ W1 + b1); probs = sigmoid(B @ W2 + b2)
    gemm128_wmma<128><<<gemmGrid, 256, 0, stream>>>(A, mW1, mb1, B, N, 1);
    mlp_out<<<(N + 7) / 8, 256, 0, stream>>>(B, mW2, mb2, out, N);
}